// KNO_DARCY_TRIANGLE_19224273617249
// MI455X (gfx1250) — compile-verified
//
#include <hip/hip_runtime.h>
#include <hip/hip_bf16.h>
#include <math.h>
#include <stdint.h>

// ---------------------------------------------------------------------------
// KNO Darcy pipeline for MI455X (gfx1250, wave32, WMMA).
// f_q = gelu(Kxq^T * Kxx^{-1} * fx) (64-RHS solve, not 1024),
// f_y = Kqy^T * Kqq^{-1} * f_q (1 RHS). Dense work on V_WMMA_F32_16X16X4_F32;
// tile staging uses GLOBAL_LOAD_ASYNC_TO_LDS_B128 (ASYNCcnt) where no
// transpose is needed, double-buffered for the K=4096 GEMM.
// ---------------------------------------------------------------------------

typedef float v2f __attribute__((ext_vector_type(2)));
typedef float v8f __attribute__((ext_vector_type(8)));

#define LDA_PAD 68   // 16 rows x stride-68 floats -> conflict-free frag reads
#define LDB_PAD 72   // row kk vs kk+2 land on disjoint bank groups

__device__ __forceinline__ float gelu_f(float x) {
  const float c0 = 0.7978845608028654f;  // sqrt(2/pi)
  const float c1 = 0.044715f;
  float x3 = x * x * x;
  return 0.5f * x * (1.0f + tanhf(c0 * (x + c1 * x3)));
}

// Async DMA: 16 bytes global -> LDS, tracked by ASYNCcnt (no VGPR data path).
__device__ __forceinline__ void async_ld_b128(uint32_t lds_addr,
                                              const float* base, uint32_t voff) {
  asm volatile("global_load_async_to_lds_b128 %0, %1, %2"
               :: "v"(lds_addr), "v"(voff), "s"(base) : "memory");
}
__device__ __forceinline__ void wait_async() {
#if __has_builtin(__builtin_amdgcn_s_wait_asynccnt)
  __builtin_amdgcn_s_wait_asynccnt(0);
#else
  asm volatile("s_wait_asynccnt 0x0" ::: "memory");
#endif
}
__device__ __forceinline__ uint32_t lds_addr_of(const void* p) {
  // generic LDS address: low 32 bits are the LDS byte offset
  return (uint32_t)(uintptr_t)p;
}

// One wave accumulates a 16x16 fp32 tile over K=64 from LDS-staged tiles.
__device__ __forceinline__ v8f wmma_k64(const float* lA, const float* lB,
                                        int mr, int nc, v8f acc) {
  int lane = threadIdx.x & 31;
  int l16  = lane & 15;
  int half = lane >> 4;
#pragma unroll
  for (int kk = 0; kk < 64; kk += 4) {
    v2f a, b;
    a.x = lA[(mr + l16) * LDA_PAD + kk + 2 * half + 0];
    a.y = lA[(mr + l16) * LDA_PAD + kk + 2 * half + 1];
    b.x = lB[(kk + 2 * half + 0) * LDB_PAD + nc + l16];
    b.y = lB[(kk + 2 * half + 1) * LDB_PAD + nc + l16];
    acc = __builtin_amdgcn_wmma_f32_16x16x4_f32(false, a, false, b,
                                                (short)0, acc, false, false);
  }
  return acc;
}

// ---------------------------------------------------------------------------
// Generic WMMA GEMM: C[M x 64] = op( A(M x K) * B(K x 64) ), M = grid.x*64,
// K multiple of 64. transA: A[m][k] = Ap[k*lda+m]. transB: B[k][n] = Bp[n*ldb+k].
// mode 0: C = act(acc + bias); mode 1: C -= acc.   (512 thr = 16 waves, 4x4)
// ---------------------------------------------------------------------------
__global__ __launch_bounds__(512) void wmma_gemm(
    const float* __restrict__ A, int lda, int transA,
    const float* __restrict__ B, int ldb, int transB,
    float* __restrict__ C, int ldc,
    const float* __restrict__ bias, int K, int mode, int act) {
  __shared__ float lA[64 * LDA_PAD];
  __shared__ float lB[64 * LDB_PAD];
  int t = threadIdx.x;
  int row0 = blockIdx.x * 64;
  int w = t >> 5, mr = (w >> 2) * 16, nc2 = (w & 3) * 16;
  v8f acc = {};
  for (int kc = 0; kc < K; kc += 64) {
    for (int idx = t; idx < 4096; idx += 512) {
      if (transA) { int i = idx & 63, k = idx >> 6;          // coalesced over i
        lA[i * LDA_PAD + k] = A[(size_t)(kc + k) * lda + row0 + i];
      } else {      int i = idx >> 6, k = idx & 63;          // coalesced over k
        lA[i * LDA_PAD + k] = A[(size_t)(row0 + i) * lda + kc + k];
      }
      if (transB) { int k = idx & 63, nn = idx >> 6;         // coalesced over k
        lB[k * LDB_PAD + nn] = B[(size_t)nn * ldb + kc + k];
      } else {      int k = idx >> 6, nn = idx & 63;         // coalesced over nn
        lB[k * LDB_PAD + nn] = B[(size_t)(kc + k) * ldb + nn];
      }
    }
    __syncthreads();
    acc = wmma_k64(lA, lB, mr, nc2, acc);
    __syncthreads();
  }
  int lane = t & 31, l16 = lane & 15, half = lane >> 4;
#pragma unroll
  for (int r = 0; r < 8; ++r) {
    int gr = row0 + mr + r + 8 * half;   // C layout: VGPR r -> rows r / r+8
    int gc = nc2 + l16;
    float v = acc[r];
    if (mode == 0) {
      if (bias) v += bias[gc];
      if (act) v = gelu_f(v);
      C[(size_t)gr * ldc + gc] = v;
    } else {
      C[(size_t)gr * ldc + gc] -= v;
    }
  }
}

// ---------------------------------------------------------------------------
// Double-buffered async-LDS WMMA GEMM (no transposes): C = gelu(A[Mx K] B[K x64])
// Prefetch chunk k+1 via GLOBAL_LOAD_ASYNC_TO_LDS_B128 while WMMAs consume k.
// Used for f_q = gelu(KxqT @ Z) with K = 4096.
// ---------------------------------------------------------------------------
__device__ __forceinline__ void stage_async_tiles(
    const float* A, int lda, int row0, int kc,
    const float* B, int ldb, const float* lAbuf, const float* lBbuf) {
  int t = threadIdx.x;
#pragma unroll
  for (int rr = 0; rr < 2; ++rr) {       // A tile: 64x64 = 1024 b128 segments
    int idx = t + rr * 512;
    int i = idx >> 4, seg = idx & 15;
    async_ld_b128(lds_addr_of(lAbuf + i * LDA_PAD + seg * 4), A,
                  (uint32_t)((((size_t)(row0 + i) * lda + kc + seg * 4)) * 4));
  }
#pragma unroll
  for (int rr = 0; rr < 2; ++rr) {       // B tile
    int idx = t + rr * 512;
    int k = idx >> 4, seg = idx & 15;
    async_ld_b128(lds_addr_of(lBbuf + k * LDB_PAD + seg * 4), B,
                  (uint32_t)((((size_t)(kc + k) * ldb + seg * 4)) * 4));
  }
}

__global__ __launch_bounds__(512) void wmma_gemm_async(
    const float* __restrict__ A, int lda,
    const float* __restrict__ B, int ldb,
    float* __restrict__ C, int ldc, int K) {
  __shared__ float lA[2][64 * LDA_PAD];
  __shared__ float lB[2][64 * LDB_PAD];
  int t = threadIdx.x;
  int row0 = blockIdx.x * 64;
  int w = t >> 5, mr = (w >> 2) * 16, nc2 = (w & 3) * 16;
  v8f acc = {};
  stage_async_tiles(A, lda, row0, 0, B, ldb, lA[0], lB[0]);
  wait_async();
  __syncthreads();
  int nch = K >> 6;
  for (int c = 0; c < nch; ++c) {
    int cur = c & 1;
    if (c + 1 < nch)
      stage_async_tiles(A, lda, row0, (c + 1) << 6, B, ldb, lA[cur ^ 1], lB[cur ^ 1]);
    acc = wmma_k64(lA[cur], lB[cur], mr, nc2, acc);
    wait_async();        // my async writes to the next buffer have landed
    __syncthreads();     // everyone done reading cur & writing next
  }
  int lane = t & 31, l16 = lane & 15, half = lane >> 4;
#pragma unroll
  for (int r = 0; r < 8; ++r) {
    int gr = row0 + mr + r + 8 * half;
    int gc = nc2 + l16;
    C[(size_t)gr * ldc + gc] = gelu_f(acc[r]);
  }
}

// ---------------------------------------------------------------------------
// Blocked Cholesky (in-place, lower triangle), block = 64
// ---------------------------------------------------------------------------
__global__ void chol_diag(float* A, int n, int kb) {
  __shared__ float L[64 * 65];
  int t = threadIdx.x;  // 64 threads
  float* Ab = A + (size_t)(kb * 64) * n + kb * 64;
  for (int r = 0; r < 64; ++r) L[r * 65 + t] = Ab[(size_t)r * n + t];
  __syncthreads();
  for (int k = 0; k < 64; ++k) {
    if (t == 0) L[k * 65 + k] = sqrtf(L[k * 65 + k]);
    __syncthreads();
    float dk = L[k * 65 + k];
    if (t > k) L[t * 65 + k] /= dk;
    __syncthreads();
    for (int j = k + 1; j < 64; ++j)
      if (t >= j) L[t * 65 + j] -= L[t * 65 + k] * L[j * 65 + k];
    __syncthreads();
  }
  for (int r = 0; r < 64; ++r) Ab[(size_t)r * n + t] = L[r * 65 + t];
}

// Panel: L[i, kb] = A[i, kb] * Lkk^{-T}, one thread per global row below block.
__global__ void chol_trsm(float* A, int n, int kb) {
  __shared__ float L[64 * 65];
  int t = threadIdx.x;
  const float* Lkk = A + (size_t)(kb * 64) * n + kb * 64;
  for (int idx = t; idx < 4096; idx += blockDim.x)
    L[(idx >> 6) * 65 + (idx & 63)] = Lkk[(size_t)(idx >> 6) * n + (idx & 63)];
  __syncthreads();
  int i = (kb + 1) * 64 + blockIdx.x * blockDim.x + t;
  if (i >= n) return;
  float* Ai = A + (size_t)i * n + kb * 64;
  float x[64];
  for (int j = 0; j < 64; ++j) {
    float s = Ai[j];
    for (int tt = 0; tt < j; ++tt) s -= x[tt] * L[j * 65 + tt];
    x[j] = s / L[j * 65 + j];
  }
  for (int j = 0; j < 64; ++j) Ai[j] = x[j];
}

// Trailing SYRK: A[ib,jb] -= P_ib * P_jb^T via WMMA; A-panel staged by async DMA.
__global__ __launch_bounds__(512) void chol_syrk(float* A, int n, int kb) {
  if (blockIdx.y > blockIdx.x) return;  // uniform per-block exit, EXEC intact
  int ib = kb + 1 + blockIdx.x, jb = kb + 1 + blockIdx.y;
  __shared__ float lA[64 * LDA_PAD];
  __shared__ float lB[64 * LDB_PAD];
  int t = threadIdx.x;
  const float* Pa = A + (size_t)(ib * 64) * n + kb * 64;
  const float* Pb = A + (size_t)(jb * 64) * n + kb * 64;
#pragma unroll
  for (int rr = 0; rr < 2; ++rr) {      // untransposed A panel: async b128 DMA
    int idx = t + rr * 512;
    int i = idx >> 4, seg = idx & 15;
    async_ld_b128(lds_addr_of(lA + i * LDA_PAD + seg * 4), Pa,
                  (uint32_t)((((size_t)i * n + seg * 4)) * 4));
  }
  for (int idx = t; idx < 4096; idx += 512) {   // B panel needs transpose
    int r = idx >> 6, c = idx & 63;
    lB[c * LDB_PAD + r] = Pb[(size_t)r * n + c];
  }
  wait_async();
  __syncthreads();
  int w = t >> 5, mr = (w >> 2) * 16, nc2 = (w & 3) * 16;
  v8f acc = {};
  acc = wmma_k64(lA, lB, mr, nc2, acc);
  int lane = t & 31, l16 = lane & 15, half = lane >> 4;
  float* Cp = A + (size_t)(ib * 64) * n + jb * 64;
#pragma unroll
  for (int r = 0; r < 8; ++r)
    Cp[(size_t)(mr + r + 8 * half) * n + nc2 + l16] -= acc[r];
}

// ---------------------------------------------------------------------------
// 64-RHS triangular diagonal-block solves (inter-block updates use wmma_gemm)
// ---------------------------------------------------------------------------
__global__ void trsm_fwd_diag(const float* A, int n, int kb, float* F) {
  __shared__ float L[64 * 65];
  int c = threadIdx.x;  // one RHS column per thread (64 threads)
  const float* Lkk = A + (size_t)(kb * 64) * n + kb * 64;
  for (int r = 0; r < 64; ++r) L[r * 65 + c] = Lkk[(size_t)r * n + c];
  __syncthreads();
  float y[64];
  float* Fb = F + (size_t)(kb * 64) * 64;
  for (int r = 0; r < 64; ++r) {
    float s = Fb[r * 64 + c];
    for (int j = 0; j < r; ++j) s -= L[r * 65 + j] * y[j];
    y[r] = s / L[r * 65 + r];
  }
  for (int r = 0; r < 64; ++r) Fb[r * 64 + c] = y[r];
}

__global__ void trsm_bwd_diag(const float* A, int n, int kb, float* F) {
  __shared__ float L[64 * 65];
  int c = threadIdx.x;
  const float* Lkk = A + (size_t)(kb * 64) * n + kb * 64;
  for (int r = 0; r < 64; ++r) L[r * 65 + c] = Lkk[(size_t)r * n + c];
  __syncthreads();
  float z[64];
  float* Fb = F + (size_t)(kb * 64) * 64;
  for (int r = 63; r >= 0; --r) {
    float s = Fb[r * 64 + c];
    for (int j = r + 1; j < 64; ++j) s -= L[j * 65 + r] * z[j];  // Lkk^T
    z[r] = s / L[r * 65 + r];
  }
  for (int r = 0; r < 64; ++r) Fb[r * 64 + c] = z[r];
}

// ---------------------------------------------------------------------------
// 1-RHS substitution for Kqq (negligible FLOPs)
// ---------------------------------------------------------------------------
__global__ void fsub1_diag(const float* A, int n, int kb, float* y) {
  if (threadIdx.x) return;
  int o = kb * 64;
  for (int r = 0; r < 64; ++r) {
    float s = y[o + r];
    for (int j = 0; j < r; ++j) s -= A[(size_t)(o + r) * n + o + j] * y[o + j];
    y[o + r] = s / A[(size_t)(o + r) * n + o + r];
  }
}
__global__ void fsub1_upd(const float* A, int n, int kb, float* y) {
  int i = (kb + 1) * 64 + blockIdx.x * blockDim.x + threadIdx.x;
  if (i >= n) return;
  int o = kb * 64; float s = 0.f;
  for (int k = 0; k < 64; ++k) s += A[(size_t)i * n + o + k] * y[o + k];
  y[i] -= s;
}
__global__ void bsub1_diag(const float* A, int n, int kb, float* y) {
  if (threadIdx.x) return;
  int o = kb * 64;
  for (int r = 63; r >= 0; --r) {
    float s = y[o + r];
    for (int j = r + 1; j < 64; ++j) s -= A[(size_t)(o + j) * n + o + r] * y[o + j];
    y[o + r] = s / A[(size_t)(o + r) * n + o + r];
  }
}
__global__ void bsub1_upd(const float* A, int n, int kb, float* y) {
  int i = blockIdx.x * blockDim.x + threadIdx.x;
  if (i >= kb * 64) return;
  int o = kb * 64; float s = 0.f;
  for (int k = 0; k < 64; ++k) s += A[(size_t)(o + k) * n + i] * y[o + k];
  y[i] -= s;
}

// ---------------------------------------------------------------------------
// Elementwise / fused kernels
// ---------------------------------------------------------------------------
__global__ void lift_kernel(const float* __restrict__ fx_in,
                            const float* __restrict__ xg,
                            const float* __restrict__ W,
                            const float* __restrict__ b,
                            float* __restrict__ FX) {
  int idx = blockIdx.x * blockDim.x + threadIdx.x;
  int m = idx >> 6, c = idx & 63;
  FX[idx] = fx_in[m] * W[c * 3 + 0] + xg[2 * m] * W[c * 3 + 1] +
            xg[2 * m + 1] * W[c * 3 + 2] + b[c];
}

__global__ void build_rbf(const float* __restrict__ X, const float* __restrict__ Y,
                          int ny, const float* __restrict__ ls,
                          float* __restrict__ K, int jitter) {
  int i = blockIdx.x;
  int j = blockIdx.y * blockDim.x + threadIdx.x;
  float l = ls[0];
  float s = 1.0f / (2.0f * l * l);
  float dx = X[2 * i] - Y[2 * j];
  float dy = X[2 * i + 1] - Y[2 * j + 1];
  float v = __expf(-(dx * dx + dy * dy) * s);
  if (jitter && i == j) v += 1e-5f;
  K[(size_t)i * ny + j] = v;
}

__global__ void build_dq(const float* __restrict__ Qn, float* __restrict__ D) {
  int i = blockIdx.x;
  int j = blockIdx.y * blockDim.x + threadIdx.x;
  float dx = Qn[2 * i] - Qn[2 * j];
  float dy = Qn[2 * i + 1] - Qn[2 * j + 1];
  D[(size_t)i * 1024 + j] = dx * dx + dy * dy;
}

// fi[k,c] = sum_q exp(-Dq[k,q]/(2 ls_c^2)) * w[q] * fq[q,c]; out = skip + fi
__global__ void integral_layer(const float* __restrict__ Dq,
                               const float* __restrict__ fq,
                               const float* __restrict__ w,
                               const float* __restrict__ skip,
                               const float* __restrict__ ls,
                               float* __restrict__ out, int act) {
  int c = threadIdx.x;   // 64 channels
  int k = blockIdx.x;    // 1024 output nodes
  float l = ls[c];
  float s = 1.0f / (2.0f * l * l);
  float acc = 0.0f;
  const float* dr = Dq + (size_t)k * 1024;
  for (int q = 0; q < 1024; ++q)
    acc += __expf(-dr[q] * s) * (w[q] * fq[q * 64 + c]);
  float v = skip[k * 64 + c] + acc;
  out[k * 64 + c] = act ? gelu_f(v) : v;
}

__global__ void p3_matvec(const float* __restrict__ fq, const float* __restrict__ W,
                          const float* __restrict__ b, float* __restrict__ out) {
  int q = blockIdx.x * blockDim.x + threadIdx.x;
  float s = b[0];
  for (int c = 0; c < 64; ++c) s += fq[q * 64 + c] * W[c];
  out[q] = s;
}

// f_y[y] = sum_q exp(-|y-q|^2/(2 out_ls^2)) * zz[q]   (Kqy never materialized)
__global__ void final_out(const float* __restrict__ yg, const float* __restrict__ qn,
                          const float* __restrict__ ls, const float* __restrict__ zz,
                          float* __restrict__ out) {
  int y = blockIdx.x * blockDim.x + threadIdx.x;
  float l = ls[0];
  float s = 1.0f / (2.0f * l * l);
  float y0 = yg[2 * y], y1 = yg[2 * y + 1];
  float acc = 0.0f;
  for (int q = 0; q < 1024; ++q) {
    float dx = y0 - qn[2 * q], dy = y1 - qn[2 * q + 1];
    acc += __expf(-(dx * dx + dy * dy) * s) * zz[q];
  }
  out[y] = acc;
}

// ---------------------------------------------------------------------------
extern "C" void kernel_launch(void* const* d_in, const int* in_sizes, int n_in,
                              void* d_out, int out_size, void* d_ws, size_t ws_size,
                              hipStream_t stream) {
  (void)in_sizes; (void)n_in; (void)out_size; (void)ws_size;
  const int N = 4096, Q = 1024, C = 64;

  const float* f_x     = (const float*)d_in[0];
  const float* x_grid  = (const float*)d_in[1];
  const float* y_grid  = (const float*)d_in[2];
  const float* q_nodes = (const float*)d_in[3];
  const float* q_w     = (const float*)d_in[4];
  const float* lift_W  = (const float*)d_in[5];
  const float* lift_b  = (const float*)d_in[6];
  const float* in_ls   = (const float*)d_in[7];
  const float* out_ls  = (const float*)d_in[8];
  const float* int_ls  = (const float*)d_in[9];
  const float* pw_W    = (const float*)d_in[10];
  const float* pw_b    = (const float*)d_in[11];
  const float* p1_W    = (const float*)d_in[12];
  const float* p1_b    = (const float*)d_in[13];
  const float* p2_W    = (const float*)d_in[14];
  const float* p2_b    = (const float*)d_in[15];
  const float* p3_W    = (const float*)d_in[16];
  const float* p3_b    = (const float*)d_in[17];
  float* out = (float*)d_out;

  // Workspace layout (floats). Kxx region is reused for Dq/Kqq once dead.
  float* ws   = (float*)d_ws;
  float* A    = ws;                    // 16,777,216  Kxx (then Dq/Kqq overlay)
  float* KxqT = ws + 16777216;         //  4,194,304
  float* FX   = ws + 20971520;         //    262,144  fx -> z (in place)
  float* FQ   = ws + 21233664;         //     65,536
  float* FQ2  = ws + 21299200;         //     65,536
  float* FQS  = ws + 21364736;         //     65,536  skip
  float* FQ3  = ws + 21430272;         //      1,024
  float* Dq   = A;                     // overlays dead Kxx
  float* Kqq  = A + 1048576;

  // 1) lift: FX = [f_x | x_grid] @ lift_W^T + b
  lift_kernel<<<N * C / 256, 256, 0, stream>>>(f_x, x_grid, lift_W, lift_b, FX);

  // 2) Kxx = rbf(x,x,in_ls) + 1e-5 I
  build_rbf<<<dim3(N, N / 256), 256, 0, stream>>>(x_grid, x_grid, N, in_ls, A, 1);

  // 3) Cholesky of Kxx (64 block steps; SYRK trailing update on fp32 WMMA)
  const int nb = N / 64;
  for (int kb = 0; kb < nb; ++kb) {
    chol_diag<<<1, 64, 0, stream>>>(A, N, kb);
    int rem = nb - kb - 1;
    if (rem) {
      chol_trsm<<<(rem * 64 + 127) / 128, 128, 0, stream>>>(A, N, kb);
      chol_syrk<<<dim3(rem, rem), 512, 0, stream>>>(A, N, kb);
    }
  }

  // 4) Forward solve L Y = FX (64 RHS), WMMA rank-64 updates below
  for (int kb = 0; kb < nb; ++kb) {
    trsm_fwd_diag<<<1, 64, 0, stream>>>(A, N, kb, FX);
    int rem = nb - kb - 1;
    if (rem)
      wmma_gemm<<<rem, 512, 0, stream>>>(A + (size_t)(kb + 1) * 64 * N + kb * 64, N, 0,
                                         FX + (size_t)kb * 64 * C, C, 0,
                                         FX + (size_t)(kb + 1) * 64 * C, C,
                                         nullptr, 64, 1, 0);
  }
  // 5) Backward solve L^T Z = Y (transposed-A WMMA updates above)
  for (int kb = nb - 1; kb >= 0; --kb) {
    trsm_bwd_diag<<<1, 64, 0, stream>>>(A, N, kb, FX);
    if (kb)
      wmma_gemm<<<kb, 512, 0, stream>>>(A + (size_t)kb * 64 * N, N, 1,
                                        FX + (size_t)kb * 64 * C, C, 0,
                                        FX, C, nullptr, 64, 1, 0);
  }

  // 6) KxqT[q][m] = rbf(q,x); f_q = gelu(KxqT @ Z)  -- async double-buffered WMMA
  build_rbf<<<dim3(Q, N / 256), 256, 0, stream>>>(q_nodes, x_grid, N, in_ls, KxqT, 0);
  wmma_gemm_async<<<Q / 64, 512, 0, stream>>>(KxqT, N, FX, C, FQ, C, N);

  // 7) Dq + DEPTH integral layers (Kxx is dead; Dq overlays it)
  build_dq<<<dim3(Q, Q / 256), 256, 0, stream>>>(q_nodes, Dq);
  float* cur = FQ; float* nxt = FQ2;
  for (int i = 0; i < 4; ++i) {
    wmma_gemm<<<Q / 64, 512, 0, stream>>>(cur, C, 0, pw_W + (size_t)i * C * C, C, 1,
                                          FQS, C, pw_b + (size_t)i * C, C, 0, 0);
    integral_layer<<<Q, 64, 0, stream>>>(Dq, cur, q_w, FQS, int_ls + (size_t)i * C,
                                         nxt, (i < 3) ? 1 : 0);
    float* tmp = cur; cur = nxt; nxt = tmp;
  }

  // 8) Projection head: p1, p2 (WMMA + fused gelu), p3 matvec
  wmma_gemm<<<Q / 64, 512, 0, stream>>>(cur, C, 0, p1_W, C, 1, nxt, C, p1_b, C, 0, 1);
  { float* tmp = cur; cur = nxt; nxt = tmp; }
  wmma_gemm<<<Q / 64, 512, 0, stream>>>(cur, C, 0, p2_W, C, 1, nxt, C, p2_b, C, 0, 1);
  { float* tmp = cur; cur = nxt; nxt = tmp; }
  p3_matvec<<<Q / 256, 256, 0, stream>>>(cur, p3_W, p3_b, FQ3);

  // 9) Kqq = rbf(q,q,out_ls)+1e-5 I; Cholesky (n=1024)
  build_rbf<<<dim3(Q, Q / 256), 256, 0, stream>>>(q_nodes, q_nodes, Q, out_ls, Kqq, 1);
  const int nb2 = Q / 64;
  for (int kb = 0; kb < nb2; ++kb) {
    chol_diag<<<1, 64, 0, stream>>>(Kqq, Q, kb);
    int rem = nb2 - kb - 1;
    if (rem) {
      chol_trsm<<<(rem * 64 + 127) / 128, 128, 0, stream>>>(Kqq, Q, kb);
      chol_syrk<<<dim3(rem, rem), 512, 0, stream>>>(Kqq, Q, kb);
    }
  }
  // 10) 1-RHS solve Kqq zz = f_q3 (in place in FQ3)
  for (int kb = 0; kb < nb2; ++kb) {
    fsub1_diag<<<1, 32, 0, stream>>>(Kqq, Q, kb, FQ3);
    int rem = nb2 - kb - 1;
    if (rem) fsub1_upd<<<(rem * 64 + 255) / 256, 256, 0, stream>>>(Kqq, Q, kb, FQ3);
  }
  for (int kb = nb2 - 1; kb >= 0; --kb) {
    bsub1_diag<<<1, 32, 0, stream>>>(Kqq, Q, kb, FQ3);
    if (kb) bsub1_upd<<<(kb * 64 + 255) / 256, 256, 0, stream>>>(Kqq, Q, kb, FQ3);
  }

  // 11) f_y = Kqy^T zz, with RBF fused into the matvec (no 16MB temporary)
  final_out<<<N / 256, 256, 0, stream>>>(y_grid, q_nodes, out_ls, FQ3, out);
}